// UnbatchAndPad_63204738728436
// MI455X (gfx1250) — compile-verified
//
#include <hip/hip_runtime.h>
#include <stdint.h>

// ---------------------------------------------------------------------------
// UnbatchAndPad for MI455X (gfx1250, wave32).
//
// batch is sorted, so the op is per-set contiguous copy + zero fill:
//   starts[b] = lower_bound(batch, b)
//   padded[b,m,:] = src[starts[b]+m,:]   if m < min(count[b], M)  else 0
//   mask[b,m]    = (m < min(count[b], M))
// Gather formulation: every output row written exactly once, no atomics.
//
// Pure bandwidth problem: ~512MB out + ~256MB in  =>  ~40us at 23.3 TB/s.
// Filled rows ride the CDNA5 async global<->LDS copy engine
// (global_load_async_to_lds_b128 / global_store_async_from_lds_b128 +
// s_wait_asynccnt); empty rows are non-temporal b128 zero stores so the
// 512MB output stream doesn't thrash the 192MB L2.
// ---------------------------------------------------------------------------

#define BATCHES 512   // B (reference constant)
#define MAXSET  1024  // M (reference constant)

typedef float v4f __attribute__((ext_vector_type(4)));
typedef int   v4i __attribute__((ext_vector_type(4)));

#if defined(__gfx1250__) &&                                        \
    __has_builtin(__builtin_amdgcn_global_load_async_to_lds_b128) && \
    __has_builtin(__builtin_amdgcn_global_store_async_from_lds_b128) && \
    __has_builtin(__builtin_amdgcn_s_wait_asynccnt)
#define USE_ASYNC_LDS 1
// Exact parameter types from clang's diagnostic: arg0 = v4i in AS(1) (global),
// arg1 = v4i in AS(3) (LDS), then two integer immediates (offset, cpol).
typedef v4i __attribute__((address_space(1))) *gv4i_ptr;
typedef v4i __attribute__((address_space(3))) *lv4i_ptr;
#else
#define USE_ASYNC_LDS 0
#endif

// ---------------------------------------------------------------------------
// Kernel A: starts[b] = lower_bound(batch, b) for b in [0, B]; starts[B] = N.
// ---------------------------------------------------------------------------
__global__ void compute_starts_kernel(const int* __restrict__ batch, int n,
                                      int* __restrict__ starts) {
    int b = blockIdx.x * blockDim.x + threadIdx.x;
    if (b > BATCHES) return;
    int lo = 0, hi = n;
    while (lo < hi) {
        int mid = (lo + hi) >> 1;
        if (batch[mid] < b) lo = mid + 1; else hi = mid;
    }
    starts[b] = lo;
}

// ---------------------------------------------------------------------------
// Kernel B: one wave32 per output row (1KB). 256-thread block = 8 waves =
// 8 rows/block. Row copy = 64 x b128; 2 iterations per lane.
// ---------------------------------------------------------------------------
__global__ __launch_bounds__(256) void gather_pad_kernel(
    const float* __restrict__ src, const int* __restrict__ starts,
    float* __restrict__ padded, float* __restrict__ mask,
    int d /* = 256 */) {

    const int lane = threadIdx.x & 31;
    const int wave = threadIdx.x >> 5;
    const long long row = (long long)blockIdx.x * 8 + wave;  // [0, B*M)
    const int b = (int)(row >> 10);        // row / M   (M = 1024)
    const int m = (int)(row & (MAXSET - 1));

    const int s0 = starts[b];
    int cnt = starts[b + 1] - s0;
    if (cnt > MAXSET) cnt = MAXSET;        // reference drops pos >= M

    const int nq = d >> 2;                 // float4s per row (64)
    v4f* __restrict__ dst4 = (v4f*)(padded + row * (long long)d);

    if (m < cnt) {
        const float* srow = src + (long long)(s0 + m) * d;
#if USE_ASYNC_LDS
        // CDNA5 path: stage the row through LDS with the async copy engine.
        // Per-wave private 1KB LDS slice; only ASYNCcnt waits needed (no
        // cross-wave sharing, so no barrier).
        __shared__ float lds_buf[8 * 256];
        float* lrow = &lds_buf[wave * 256];
        for (int i = lane; i < nq; i += 32) {
            __builtin_amdgcn_global_load_async_to_lds_b128(
                (gv4i_ptr)(srow + 4 * i),
                (lv4i_ptr)(lrow + 4 * i),
                0, 0);
        }
        __builtin_amdgcn_s_wait_asynccnt(0);   // LDS now holds the row
        for (int i = lane; i < nq; i += 32) {
            __builtin_amdgcn_global_store_async_from_lds_b128(
                (gv4i_ptr)((float*)dst4 + 4 * i),
                (lv4i_ptr)(lrow + 4 * i),
                0, 0);
        }
        __builtin_amdgcn_s_wait_asynccnt(0);   // LDS slice reusable / done
#else
        // Fallback: b128 register copy, non-temporal both ways (streamed once).
        const v4f* __restrict__ s4 = (const v4f*)srow;
        for (int i = lane; i < nq; i += 32) {
            v4f v = __builtin_nontemporal_load(s4 + i);
            __builtin_nontemporal_store(v, dst4 + i);
        }
#endif
    } else {
        // Zero fill: non-temporal b128 stores.
        v4f z = {0.0f, 0.0f, 0.0f, 0.0f};
        for (int i = lane; i < nq; i += 32) {
            __builtin_nontemporal_store(z, dst4 + i);
        }
    }

    if (lane == 0) {
        mask[row] = (m < cnt) ? 1.0f : 0.0f;
    }
}

// ---------------------------------------------------------------------------
// Launch. Inputs: [0] src (N*D f32), [1] batch (N int, sorted),
//                 [2] num_sets (1), [3] max_set_size (1).
// d_out: padded (B*M*D f32) then mask (B*M) flattened.
// d_ws: starts[B+1] ints.
// ---------------------------------------------------------------------------
extern "C" void kernel_launch(void* const* d_in, const int* in_sizes, int n_in,
                              void* d_out, int out_size, void* d_ws, size_t ws_size,
                              hipStream_t stream) {
    const float* src   = (const float*)d_in[0];
    const int*   batch = (const int*)d_in[1];

    const int n = in_sizes[1];                 // 262144
    const int d = in_sizes[0] / n;             // 256

    int* starts = (int*)d_ws;                  // B+1 ints of scratch

    compute_starts_kernel<<<(BATCHES + 1 + 255) / 256, 256, 0, stream>>>(
        batch, n, starts);

    float* padded = (float*)d_out;
    float* mask   = padded + (long long)BATCHES * MAXSET * d;

    const int rows = BATCHES * MAXSET;         // 524288 rows, 8 per block
    gather_pad_kernel<<<rows / 8, 256, 0, stream>>>(src, starts, padded, mask, d);
}